// INT4SymmetricWeightsDecompressor_10187662426742
// MI455X (gfx1250) — compile-verified
//
#include <hip/hip_runtime.h>
#include <hip/hip_bf16.h>

// INT4 symmetric grouped dequantization (LLaMA-7B FFN linear weights)
//   OUT=4096 rows, GROUPS=86, GSIZE=128  -> output (4096, 11008) fp32
//   packed: int32[TOTAL/2], each element holds one byte = two 4-bit codes
//   scale : float32[OUT*GROUPS]
// Memory-bound: ~272 MB traffic, AI ~0.33 flop/byte -> pure streaming kernel.

#define OUT_ROWS 4096
#define GROUPS   86
#define GSIZE    128
#define TOTAL    (OUT_ROWS * GROUPS * GSIZE)        // 45,088,768 outputs
#define NPACKED  (TOTAL / 2)                        // 22,544,384 packed int32

// 16 outputs per thread => 8 packed words per thread
#define OUT_PER_THREAD 16
#define BLOCK 256

typedef int   v4i __attribute__((ext_vector_type(4)));
typedef float v4f __attribute__((ext_vector_type(4)));

__global__ __launch_bounds__(BLOCK) void int4_dequant_kernel(
    const int*   __restrict__ packed,
    const float* __restrict__ scale,
    float*       __restrict__ out)
{
    const unsigned tid   = blockIdx.x * BLOCK + threadIdx.x;      // < TOTAL/16 exactly
    const unsigned pbase = tid * 8u;                              // 8 packed words / thread
    const unsigned obase = tid * 16u;                             // 16 fp32 outputs / thread

    // Speculative prefetch ~64KB ahead to warm L2 for later blocks
    // (emits global_prefetch_b8; OOB speculative prefetch is dropped by HW).
    __builtin_prefetch(packed + pbase + 16384, 0, 0);

    // Non-temporal b128 loads: packed stream is read exactly once.
    const v4i p0 = __builtin_nontemporal_load((const v4i*)(packed + pbase));
    const v4i p1 = __builtin_nontemporal_load((const v4i*)(packed + pbase + 4));

    // Flat group index of output e is e/128 (row stride 11008 = 86*128),
    // so for a 16-output chunk: scale_idx = (tid*16)>>7 = tid>>3.
    // 8 consecutive lanes share one scale -> broadcast-friendly.
    const float s = scale[tid >> 3];
    const float nb = -8.0f * s;   // exact (power-of-2 scale of s)

    // Dequant: (code - 8) * s  computed as  fma(code, s, -8*s)
    // -> single rounding, bit-exact vs. reference fl((code-8)*s).
    float r[16];
#pragma unroll
    for (int k = 0; k < 4; ++k) {
        const int v = p0[k];
        r[2 * k + 0] = __builtin_fmaf((float)(v & 15),        s, nb);
        r[2 * k + 1] = __builtin_fmaf((float)((v >> 4) & 15), s, nb);
    }
#pragma unroll
    for (int k = 0; k < 4; ++k) {
        const int v = p1[k];
        r[8 + 2 * k + 0] = __builtin_fmaf((float)(v & 15),        s, nb);
        r[8 + 2 * k + 1] = __builtin_fmaf((float)((v >> 4) & 15), s, nb);
    }

    // Non-temporal b128 stores: 180 MB written once, never re-read ->
    // keep it from displacing useful lines in the 192 MB L2.
#pragma unroll
    for (int q = 0; q < 4; ++q) {
        v4f vv;
        vv.x = r[4 * q + 0];
        vv.y = r[4 * q + 1];
        vv.z = r[4 * q + 2];
        vv.w = r[4 * q + 3];
        __builtin_nontemporal_store(vv, (v4f*)(out + obase + 4 * q));
    }
}

extern "C" void kernel_launch(void* const* d_in, const int* in_sizes, int n_in,
                              void* d_out, int out_size, void* d_ws, size_t ws_size,
                              hipStream_t stream)
{
    const int*   packed = (const int*)d_in[0];     // int32, one byte payload each
    const float* scale  = (const float*)d_in[1];   // float32 (4096, 86, 1) flat
    float*       out    = (float*)d_out;           // float32 (4096, 11008) flat

    const unsigned nthreads = TOTAL / OUT_PER_THREAD;   // 2,818,048
    const unsigned nblocks  = nthreads / BLOCK;         // 11,008 (exact)

    int4_dequant_kernel<<<nblocks, BLOCK, 0, stream>>>(packed, scale, out);
}